// LocalTransformerBlock_9809705304973
// MI455X (gfx1250) — compile-verified
//
#include <hip/hip_runtime.h>
#include <math.h>

#define HID 128
#define EDGE_D 32
#define HEADS 8
#define DH 16

typedef __attribute__((ext_vector_type(2))) float v2f;
typedef __attribute__((ext_vector_type(8))) float v8f;

// ---------------------------------------------------------------------------
// f32 WMMA: D(16x16) = A(16x4) * B(4x16) + C   (V_WMMA_F32_16X16X4_F32)
// A frag: lane&15 = row M; (lane>>4)*2 = K base; {v0,v1} = K, K+1  -> float2
// B frag: lane&15 = col N; (lane>>4)*2 = K base; {v0,v1} = rows K, K+1
// C/D   : vgpr r -> M = r + 8*(lane>>4), N = lane&15
// ---------------------------------------------------------------------------
__device__ __forceinline__ v8f wmma_f32_k4(v2f a, v2f b, v8f c) {
  return __builtin_amdgcn_wmma_f32_16x16x4_f32(false, a, false, b, (short)0, c,
                                               false, false);
}

__device__ __forceinline__ void atomicMaxF(float* addr, float val) {
  int* ai = (int*)addr;
  int old = *ai;
  while (__int_as_float(old) < val) {
    int assumed = old;
    old = atomicCAS(ai, assumed, __float_as_int(val));
    if (old == assumed) break;
  }
}

// ---------------------------------------------------------------------------
// init: amax=-inf, denom=0, msgnum=0
// ---------------------------------------------------------------------------
__global__ void init_kernel(float* __restrict__ amax, float* __restrict__ denom,
                            float* __restrict__ msgnum, int n8, int n128) {
  int i = blockIdx.x * blockDim.x + threadIdx.x;
  if (i < n8) { amax[i] = -INFINITY; denom[i] = 0.0f; }
  if (i < n128) msgnum[i] = 0.0f;
}

// ---------------------------------------------------------------------------
// LayerNorm over HID=128, one wave32 per row, 8 rows per 256-thread block
// ---------------------------------------------------------------------------
__global__ __launch_bounds__(256)
void ln_kernel(const float* __restrict__ x, const float* __restrict__ w,
               const float* __restrict__ b, float* __restrict__ out, int n) {
  int wave = threadIdx.x >> 5;
  int lane = threadIdx.x & 31;
  int row = blockIdx.x * 8 + wave;
  if (row >= n) return;
  const float* xr = x + (size_t)row * HID;
  float4 val = *(const float4*)(xr + lane * 4);
  float s  = val.x + val.y + val.z + val.w;
  float ss = val.x * val.x + val.y * val.y + val.z * val.z + val.w * val.w;
  #pragma unroll
  for (int off = 16; off > 0; off >>= 1) {
    s  += __shfl_xor(s, off, 32);
    ss += __shfl_xor(ss, off, 32);
  }
  float mean = s * (1.0f / HID);
  float var  = ss * (1.0f / HID) - mean * mean;
  float inv  = rsqrtf(var + 1e-5f);
  float4 wv = *(const float4*)(w + lane * 4);
  float4 bv = *(const float4*)(b + lane * 4);
  float4 o;
  o.x = (val.x - mean) * inv * wv.x + bv.x;
  o.y = (val.y - mean) * inv * wv.y + bv.y;
  o.z = (val.z - mean) * inv * wv.z + bv.z;
  o.w = (val.w - mean) * inv * wv.w + bv.w;
  *(float4*)(out + (size_t)row * HID + lane * 4) = o;
}

// ---------------------------------------------------------------------------
// q/k/v/skip = h @ {Wq,Wk,Wv,Wskip} + bias.  One wave per 16x16 tile.
// grid = (N/16, 4), block = 256 (8 waves -> 8 col tiles; 32 total across y)
// ---------------------------------------------------------------------------
__global__ __launch_bounds__(256)
void gemm_qkvs_kernel(const float* __restrict__ h,
                      const float* __restrict__ Wq, const float* __restrict__ bq,
                      const float* __restrict__ Wk, const float* __restrict__ bk,
                      const float* __restrict__ Wv, const float* __restrict__ bv,
                      const float* __restrict__ Ws, const float* __restrict__ bs,
                      float* __restrict__ q, float* __restrict__ k,
                      float* __restrict__ v, float* __restrict__ s) {
  int wave = threadIdx.x >> 5;
  int lane = threadIdx.x & 31;
  int m0 = blockIdx.x * 16;
  int ct = blockIdx.y * 8 + wave;      // 0..31
  int mat = ct >> 3;
  int n0 = (ct & 7) * 16;
  const float* W; const float* bias; float* out;
  if (mat == 0)      { W = Wq; bias = bq; out = q; }
  else if (mat == 1) { W = Wk; bias = bk; out = k; }
  else if (mat == 2) { W = Wv; bias = bv; out = v; }
  else               { W = Ws; bias = bs; out = s; }

  int col = lane & 15;                 // also A row (same lane mapping)
  int kh  = (lane >> 4) << 1;
  v8f c;
  float bb = bias[n0 + col];
  #pragma unroll
  for (int r = 0; r < 8; ++r) c[r] = bb;
  const float* Arow = h + (size_t)(m0 + col) * HID;
  #pragma unroll 4
  for (int kk = 0; kk < HID; kk += 4) {
    v2f a = *(const v2f*)(Arow + kk + kh);
    v2f b;
    b.x = W[(kk + kh) * HID + n0 + col];
    b.y = W[(kk + kh + 1) * HID + n0 + col];
    c = wmma_f32_k4(a, b, c);
  }
  int rbase = m0 + ((lane >> 4) << 3);
  #pragma unroll
  for (int r = 0; r < 8; ++r)
    out[(size_t)(rbase + r) * HID + n0 + col] = c[r];
}

// ---------------------------------------------------------------------------
// Recompute e = edge_attr@We + be for 64 edges into LDS via WMMA.
// Waves 0..3 each own a 16-edge strip (8 col-tiles, K=32 -> 8 WMMAs each).
// ---------------------------------------------------------------------------
__device__ __forceinline__ void e_tiles_to_lds(const float* __restrict__ edge_attr,
                                               const float* __restrict__ We,
                                               const float* __restrict__ be,
                                               long long e0, int wave, int lane,
                                               float* __restrict__ lds_e) {
  if (wave < 4) {
    int col = lane & 15;
    int kh  = (lane >> 4) << 1;
    const float* A = edge_attr + (size_t)(e0 + wave * 16) * EDGE_D;
    int rbase = wave * 16 + ((lane >> 4) << 3);
    #pragma unroll
    for (int ct = 0; ct < 8; ++ct) {
      int n0 = ct * 16;
      v8f c;
      float bb = be[n0 + col];
      #pragma unroll
      for (int r = 0; r < 8; ++r) c[r] = bb;
      #pragma unroll
      for (int kk = 0; kk < EDGE_D; kk += 4) {
        v2f a = *(const v2f*)(A + col * EDGE_D + kk + kh);
        v2f b;
        b.x = We[(kk + kh) * HID + n0 + col];
        b.y = We[(kk + kh + 1) * HID + n0 + col];
        c = wmma_f32_k4(a, b, c);
      }
      #pragma unroll
      for (int r = 0; r < 8; ++r)
        lds_e[(rbase + r) * HID + n0 + col] = c[r];
    }
  }
  __syncthreads();
}

// ---------------------------------------------------------------------------
// Edge pass 1: alpha = q[dst].(k[src]+e)/4 per head; amax via atomic max
// 64 edges/block; 4 threads per edge (2 heads each)
// ---------------------------------------------------------------------------
__global__ __launch_bounds__(256)
void edge_alpha_kernel(const float* __restrict__ edge_attr,
                       const float* __restrict__ We, const float* __restrict__ be,
                       const int* __restrict__ ei, const float* __restrict__ q,
                       const float* __restrict__ k, float* __restrict__ alpha,
                       float* __restrict__ amax, int E) {
  __shared__ float lds_e[64 * HID];
  long long e0 = (long long)blockIdx.x * 64;
  int wave = threadIdx.x >> 5, lane = threadIdx.x & 31;
  e_tiles_to_lds(edge_attr, We, be, e0, wave, lane, lds_e);

  int le = threadIdx.x >> 2;           // local edge 0..63
  int h0 = (threadIdx.x & 3) * 2;
  long long e = e0 + le;
  int src = ei[e];
  int dst = ei[(long long)E + e];
  const float* qrow = q + (size_t)dst * HID;
  const float* krow = k + (size_t)src * HID;
  const float* erow = lds_e + le * HID;
  #pragma unroll
  for (int h = h0; h < h0 + 2; ++h) {
    float acc = 0.0f;
    #pragma unroll
    for (int d = 0; d < DH; ++d)
      acc += qrow[h * DH + d] * (krow[h * DH + d] + erow[h * DH + d]);
    acc *= 0.25f;                      // 1/sqrt(16)
    alpha[e * HEADS + h] = acc;
    atomicMaxF(&amax[(size_t)dst * HEADS + h], acc);
  }
}

// ---------------------------------------------------------------------------
// Edge pass 2: expa = exp(alpha - amax[dst]); denom += expa;
//              msgnum[dst] += expa * (v[src] + e)
// ---------------------------------------------------------------------------
__global__ __launch_bounds__(256)
void edge_msg_kernel(const float* __restrict__ edge_attr,
                     const float* __restrict__ We, const float* __restrict__ be,
                     const int* __restrict__ ei, const float* __restrict__ v,
                     const float* __restrict__ alpha, const float* __restrict__ amax,
                     float* __restrict__ denom, float* __restrict__ msgnum, int E) {
  __shared__ float lds_e[64 * HID];
  long long e0 = (long long)blockIdx.x * 64;
  int wave = threadIdx.x >> 5, lane = threadIdx.x & 31;
  e_tiles_to_lds(edge_attr, We, be, e0, wave, lane, lds_e);

  int le = threadIdx.x >> 2;
  int h0 = (threadIdx.x & 3) * 2;
  long long e = e0 + le;
  int src = ei[e];
  int dst = ei[(long long)E + e];
  const float* vrow = v + (size_t)src * HID;
  const float* erow = lds_e + le * HID;
  float* mrow = msgnum + (size_t)dst * HID;
  #pragma unroll
  for (int h = h0; h < h0 + 2; ++h) {
    float ex = expf(alpha[e * HEADS + h] - amax[(size_t)dst * HEADS + h]);
    atomicAdd(&denom[(size_t)dst * HEADS + h], ex);
    #pragma unroll
    for (int d = 0; d < DH; ++d)
      atomicAdd(&mrow[h * DH + d], ex * (vrow[h * DH + d] + erow[h * DH + d]));
  }
}

// ---------------------------------------------------------------------------
// xnew = x + msgnum/(denom+1e-16) + skip
// ---------------------------------------------------------------------------
__global__ void node_update_kernel(const float* __restrict__ x,
                                   const float* __restrict__ msgnum,
                                   const float* __restrict__ denom,
                                   const float* __restrict__ skip,
                                   float* __restrict__ xnew, int total) {
  int i = blockIdx.x * blockDim.x + threadIdx.x;
  if (i >= total) return;
  int node = i >> 7;
  int h = (i & 127) >> 4;
  float msg = msgnum[i] / (denom[(size_t)node * HEADS + h] + 1e-16f);
  xnew[i] = x[i] + msg + skip[i];
}

// ---------------------------------------------------------------------------
// FFN1: g = gelu(h2 @ W1 + b1)   [N,128]@[128,512]
// ---------------------------------------------------------------------------
__global__ __launch_bounds__(256)
void ffn1_kernel(const float* __restrict__ h2, const float* __restrict__ W1,
                 const float* __restrict__ b1, float* __restrict__ g) {
  int wave = threadIdx.x >> 5, lane = threadIdx.x & 31;
  int m0 = blockIdx.x * 16;
  int ct = blockIdx.y * 8 + wave;      // 0..31
  int n0 = ct * 16;
  int col = lane & 15;
  int kh  = (lane >> 4) << 1;
  v8f c;
  float bb = b1[n0 + col];
  #pragma unroll
  for (int r = 0; r < 8; ++r) c[r] = bb;
  const float* Arow = h2 + (size_t)(m0 + col) * HID;
  #pragma unroll 4
  for (int kk = 0; kk < HID; kk += 4) {
    v2f a = *(const v2f*)(Arow + kk + kh);
    v2f b;
    b.x = W1[(kk + kh) * 512 + n0 + col];
    b.y = W1[(kk + kh + 1) * 512 + n0 + col];
    c = wmma_f32_k4(a, b, c);
  }
  int rbase = m0 + ((lane >> 4) << 3);
  #pragma unroll
  for (int r = 0; r < 8; ++r) {
    float t = c[r];
    g[(size_t)(rbase + r) * 512 + n0 + col] =
        0.5f * t * (1.0f + erff(t * 0.70710678118654752f));
  }
}

// ---------------------------------------------------------------------------
// FFN2 + residual: out = xnew + g @ W2 + b2   [N,512]@[512,128]
// ---------------------------------------------------------------------------
__global__ __launch_bounds__(256)
void ffn2_kernel(const float* __restrict__ g, const float* __restrict__ W2,
                 const float* __restrict__ b2, const float* __restrict__ xnew,
                 float* __restrict__ out) {
  int wave = threadIdx.x >> 5, lane = threadIdx.x & 31;
  int m0 = blockIdx.x * 16;
  int n0 = wave * 16;                  // 8 col tiles of 128
  int col = lane & 15;
  int kh  = (lane >> 4) << 1;
  v8f c;
  float bb = b2[n0 + col];
  #pragma unroll
  for (int r = 0; r < 8; ++r) c[r] = bb;
  const float* Arow = g + (size_t)(m0 + col) * 512;
  #pragma unroll 4
  for (int kk = 0; kk < 512; kk += 4) {
    v2f a = *(const v2f*)(Arow + kk + kh);
    v2f b;
    b.x = W2[(kk + kh) * HID + n0 + col];
    b.y = W2[(kk + kh + 1) * HID + n0 + col];
    c = wmma_f32_k4(a, b, c);
  }
  int rbase = m0 + ((lane >> 4) << 3);
  #pragma unroll
  for (int r = 0; r < 8; ++r) {
    size_t idx = (size_t)(rbase + r) * HID + n0 + col;
    out[idx] = xnew[idx] + c[r];
  }
}

// ---------------------------------------------------------------------------
extern "C" void kernel_launch(void* const* d_in, const int* in_sizes, int n_in,
                              void* d_out, int out_size, void* d_ws, size_t ws_size,
                              hipStream_t stream) {
  const float* x         = (const float*)d_in[0];
  const int*   ei        = (const int*)d_in[1];
  const float* edge_attr = (const float*)d_in[2];
  const float* Wq = (const float*)d_in[3];  const float* bq = (const float*)d_in[4];
  const float* Wk = (const float*)d_in[5];  const float* bk = (const float*)d_in[6];
  const float* Wv = (const float*)d_in[7];  const float* bv = (const float*)d_in[8];
  const float* We = (const float*)d_in[9];  const float* be = (const float*)d_in[10];
  const float* Ws = (const float*)d_in[11]; const float* bs = (const float*)d_in[12];
  const float* ln1w = (const float*)d_in[13]; const float* ln1b = (const float*)d_in[14];
  const float* W1 = (const float*)d_in[15]; const float* b1 = (const float*)d_in[16];
  const float* W2 = (const float*)d_in[17]; const float* b2 = (const float*)d_in[18];
  const float* ln2w = (const float*)d_in[19]; const float* ln2b = (const float*)d_in[20];

  const int N = in_sizes[0] / HID;
  const int E = in_sizes[2] / EDGE_D;

  float* ws = (float*)d_ws;
  float* h      = ws;                      // N*128
  float* q      = h + (size_t)N * HID;     // N*128
  float* k      = q + (size_t)N * HID;     // N*128
  float* v      = k + (size_t)N * HID;     // N*128
  float* skip   = v + (size_t)N * HID;     // N*128
  float* alpha  = skip + (size_t)N * HID;  // E*8
  float* amax   = alpha + (size_t)E * HEADS;   // N*8
  float* denom  = amax + (size_t)N * HEADS;    // N*8
  float* msgnum = denom + (size_t)N * HEADS;   // N*128
  float* xnew   = msgnum + (size_t)N * HID;    // N*128
  float* g      = q;   // FFN intermediate reuses q..skip (contiguous N*512)

  const int mtiles = N / 16;               // 50000/16 = 3125
  const int eblocks = E / 64;              // 800000/64 = 12500
  const int n128 = N * HID;

  // 0) init reduction buffers
  init_kernel<<<(n128 + 255) / 256, 256, 0, stream>>>(amax, denom, msgnum,
                                                      N * HEADS, n128);
  // 1) h = LN1(x)
  ln_kernel<<<(N + 7) / 8, 256, 0, stream>>>(x, ln1w, ln1b, h, N);
  // 2) q,k,v,skip projections (WMMA)
  gemm_qkvs_kernel<<<dim3(mtiles, 4), 256, 0, stream>>>(
      h, Wq, bq, Wk, bk, Wv, bv, Ws, bs, q, k, v, skip);
  // 3) attention logits + segment max
  edge_alpha_kernel<<<eblocks, 256, 0, stream>>>(edge_attr, We, be, ei, q, k,
                                                 alpha, amax, E);
  // 4) exp / denom / weighted message accumulation
  edge_msg_kernel<<<eblocks, 256, 0, stream>>>(edge_attr, We, be, ei, v, alpha,
                                               amax, denom, msgnum, E);
  // 5) xnew = x + msg + skip
  node_update_kernel<<<(n128 + 255) / 256, 256, 0, stream>>>(x, msgnum, denom,
                                                             skip, xnew, n128);
  // 6) h2 = LN2(xnew)  (reuse h)
  ln_kernel<<<(N + 7) / 8, 256, 0, stream>>>(xnew, ln2w, ln2b, h, N);
  // 7) g = gelu(h2 @ W1 + b1)
  ffn1_kernel<<<dim3(mtiles, 4), 256, 0, stream>>>(h, W1, b1, g);
  // 8) out = xnew + g @ W2 + b2
  ffn2_kernel<<<mtiles, 256, 0, stream>>>(g, W2, b2, xnew, (float*)d_out);
}